// AdaptiveTopKSelector_8495445311708
// MI455X (gfx1250) — compile-verified
//
#include <hip/hip_runtime.h>
#include <hip/hip_bf16.h>
#include <stdint.h>

// ---------------------------------------------------------------------------
// Causal top-k selector for MI455X (gfx1250).
//   - TDM (tensor_load_to_lds) stages each 16KB row into LDS asynchronously
//   - 4-pass MSB-first radix select over order-preserving uint keys
//   - parallel descending-scan digit selection (no serial 256-bin walk)
//   - deterministic index-ascending tie break via LDS block scans
//   - non-temporal b128 output stores (streamed, never re-read)
// Memory-bound: ~550MB of HBM traffic -> ~24us floor @ 23.3 TB/s.
// ---------------------------------------------------------------------------

#define NT 256               // threads per block (8 wave32s)
#define SK 4096              // row length (Sk == Sq, power of two)
#define EPT (SK / NT)        // 16 contiguous columns per thread
#define NEG_INF_F (-1e9f)

typedef __attribute__((ext_vector_type(4))) unsigned int u32x4;
typedef __attribute__((ext_vector_type(8))) int          i32x8;
typedef __attribute__((ext_vector_type(4))) int          i32x4;
typedef __attribute__((ext_vector_type(4))) float        f32x4;  // native vector: OK for nontemporal builtin

#if __has_builtin(__builtin_amdgcn_tensor_load_to_lds)
#define HAVE_TDM 1
#else
#define HAVE_TDM 0
#endif

__device__ __forceinline__ uint32_t f2ord(float f) {
  // Monotonic float -> uint mapping: larger float => larger uint.
  uint32_t u = __float_as_uint(f);
  return u ^ ((uint32_t)((int32_t)u >> 31) | 0x80000000u);
}

struct Smem {
  uint32_t keys[SK];   // 16 KB row staging / keys (TDM target)
  uint32_t hist[256];  // radix histogram
  uint32_t sa[NT];     // scan ping
  uint32_t sb[NT];     // scan pong
  uint32_t bcast[2];   // {prefix, krem} broadcast
};

// Block-wide exclusive scan (Hillis-Steele, ping-pong in LDS).
__device__ __forceinline__ uint32_t block_excl_scan(uint32_t v, uint32_t* a,
                                                    uint32_t* b, int tid) {
  a[tid] = v;
  __syncthreads();
  uint32_t* src = a;
  uint32_t* dst = b;
#pragma unroll
  for (int off = 1; off < NT; off <<= 1) {
    uint32_t x = src[tid];
    if (tid >= off) x += src[tid - off];
    dst[tid] = x;
    __syncthreads();
    uint32_t* t = src; src = dst; dst = t;
  }
  return src[tid] - v;  // exclusive
}

__global__ void __launch_bounds__(NT)
topk_causal_kernel(const float* __restrict__ scores,
                   float* __restrict__ mask_out,
                   float* __restrict__ idx_out,
                   float* __restrict__ sparsity_out,
                   int K) {
  __shared__ Smem sm;
  const int tid = (int)threadIdx.x;
  const int row = (int)blockIdx.x;      // row = b*Sq + q
  const int q   = row & (SK - 1);       // Sq == SK is a power of two: no sdiv

  // ---- 1) Stage the row into LDS via the Tensor Data Mover ----------------
#if HAVE_TDM
  if (tid < 32) {  // wave 0 issues the DMA; TDM ignores EXEC within the wave
    const uint64_t gaddr = (uint64_t)(uintptr_t)(scores + (size_t)row * SK);
    // Low 32 bits of a generic shared pointer are the LDS byte address.
    const uint32_t lds_addr = (uint32_t)(uintptr_t)(&sm.keys[0]);

    u32x4 g0;
    g0[0] = 1u;                                   // count=1, is_restore=0
    g0[1] = lds_addr;                             // lds_addr (bytes)
    g0[2] = (uint32_t)(gaddr & 0xFFFFFFFFu);      // global_addr[31:0]
    g0[3] = (uint32_t)((gaddr >> 32) & 0x01FFFFFFu) | (2u << 30);  // [56:32] + type=2

    i32x8 g1;
    g1[0] = (int)(2u << 16);                      // data_size=4B, wg_mask=0
    g1[1] = (int)((uint32_t)SK << 16);            // tensor_dim0[15:0]
    g1[2] = (int)(1u << 16);                      // tensor_dim0[31:16]=0, tensor_dim1=1
    g1[3] = (int)((uint32_t)SK << 16);            // tile_dim0 = SK
    g1[4] = 1;                                    // tile_dim1=1, tile_dim2=0
    g1[5] = SK;                                   // tensor_dim0_stride[31:0]
    g1[6] = 0;
    g1[7] = 0;

    i32x4 z4 = {0, 0, 0, 0};
#if defined(__clang_major__) && (__clang_major__ >= 23)
    i32x8 z8 = {0, 0, 0, 0, 0, 0, 0, 0};
    __builtin_amdgcn_tensor_load_to_lds(g0, g1, z4, z4, z8, 0);
#else
    __builtin_amdgcn_tensor_load_to_lds(g0, g1, z4, z4, 0);
#endif
    __builtin_amdgcn_s_wait_tensorcnt(0);
  }
  __syncthreads();
  // Causal mask + key transform, in place over the DMA'd bits.
  for (int i = tid; i < SK; i += NT) {
    float v = __uint_as_float(sm.keys[i]);
    if (i > q) v = NEG_INF_F;
    sm.keys[i] = f2ord(v);
  }
#else
  for (int i = tid; i < SK; i += NT) {
    float v = scores[(size_t)row * SK + i];
    if (i > q) v = NEG_INF_F;
    sm.keys[i] = f2ord(v);
  }
#endif
  __syncthreads();

  // ---- 2) 4-pass MSB-first radix select: key of K-th largest --------------
  uint32_t prefix = 0;
  uint32_t krem   = (uint32_t)K;  // elements still needed within prefix group
#pragma unroll 1
  for (int pass = 0; pass < 4; ++pass) {
    const int shift = 24 - 8 * pass;
    sm.hist[tid] = 0;  // NT == 256 bins
    __syncthreads();
    const uint32_t pmask = pass ? (0xFFFFFFFFu << (shift + 8)) : 0u;
    for (int i = tid; i < SK; i += NT) {
      const uint32_t kk = sm.keys[i];
      if ((kk & pmask) == prefix)
        atomicAdd(&sm.hist[(kk >> shift) & 0xFFu], 1u);
    }
    __syncthreads();

    // Parallel digit selection: thread t owns digit d = 255 - t.
    // Exclusive scan of hist in descending-digit order gives, per digit,
    // the count of keys in strictly-greater bins within the prefix group.
    const int d = 255 - tid;
    const uint32_t cnt = sm.hist[d];
    const uint32_t cumGreater = block_excl_scan(cnt, sm.sa, sm.sb, tid);
    if (cumGreater < krem && krem <= cumGreater + cnt) {  // exactly one thread
      sm.bcast[0] = prefix | ((uint32_t)d << shift);
      sm.bcast[1] = krem - cumGreater;
    }
    __syncthreads();
    prefix = sm.bcast[0];
    krem   = sm.bcast[1];
  }
  const uint32_t T = prefix;  // key value of the K-th largest element
  const uint32_t r = krem;    // how many ==T ties to accept (lowest index first)

  // ---- 3) Selection with deterministic index-ascending tie break ----------
  const int base = tid * EPT;

  uint32_t eqc = 0;
#pragma unroll
  for (int j = 0; j < EPT; ++j) eqc += (sm.keys[base + j] == T) ? 1u : 0u;
  const uint32_t eqBefore = block_excl_scan(eqc, sm.sa, sm.sb, tid);

  // Mask output (coalesced: each wave writes a dense 2KB span, NT b128 stores)
  uint32_t selc = 0;
  {
    uint32_t eqrun = eqBefore;
    f32x4* mrow = reinterpret_cast<f32x4*>(mask_out + (size_t)row * SK) +
                  (size_t)tid * (EPT / 4);
#pragma unroll
    for (int v4 = 0; v4 < EPT / 4; ++v4) {
      f32x4 m;
#pragma unroll
      for (int j = 0; j < 4; ++j) {
        const uint32_t kk = sm.keys[base + v4 * 4 + j];
        bool sel;
        if (kk > T)       sel = true;
        else if (kk == T) { sel = (eqrun < r); ++eqrun; }
        else              sel = false;
        selc += sel ? 1u : 0u;
        m[j] = sel ? 1.0f : 0.0f;
      }
      __builtin_nontemporal_store(m, &mrow[v4]);  // streamed output, skip L2 rinse
    }
  }
  __syncthreads();
  const uint32_t selBefore = block_excl_scan(selc, sm.sa, sm.sb, tid);

  // Index output: the K selected columns, ascending column order (deterministic)
  {
    uint32_t eqrun = eqBefore;
    uint32_t pos   = selBefore;
    float* irow = idx_out + (size_t)row * (size_t)K;
#pragma unroll
    for (int j = 0; j < EPT; ++j) {
      const int col = base + j;
      const uint32_t kk = sm.keys[col];
      bool sel;
      if (kk > T)       sel = true;
      else if (kk == T) { sel = (eqrun < r); ++eqrun; }
      else              sel = false;
      if (sel && pos < (uint32_t)K) {
        __builtin_nontemporal_store((float)col, &irow[pos]);
        ++pos;
      }
    }
  }

  if (row == 0 && tid == 0)
    sparsity_out[0] = 1.0f - (float)K / (float)SK;
}

extern "C" void kernel_launch(void* const* d_in, const int* in_sizes, int n_in,
                              void* d_out, int out_size, void* d_ws, size_t ws_size,
                              hipStream_t stream) {
  (void)n_in; (void)d_ws; (void)ws_size;
  const float* scores = (const float*)d_in[0];
  // d_in[1] is top_k; K is recoverable exactly from out_size, so no host read.

  const size_t total = (size_t)in_sizes[0];      // B*Sq*Sk
  const int rows = (int)(total / SK);            // B*Sq
  const int K    = (int)(((size_t)out_size - total - 1) / (size_t)rows);  // = 512

  float* mask = (float*)d_out;
  float* idx  = mask + total;
  float* sp   = idx + (size_t)rows * (size_t)K;

  topk_causal_kernel<<<rows, NT, 0, stream>>>(scores, mask, idx, sp, K);
}